// DWT3D_80418967650974
// MI455X (gfx1250) — compile-verified
//
#include <hip/hip_runtime.h>

// 3-D Haar DWT (stride-2 2x2x2 grouped conv) as WMMA f32 16x16x4 matmuls.
// A = diag(F, F) where F is the 8x8 Haar matrix, F[s][k] = c * (-1)^popcount(s&k),
// c = (1/sqrt(2))^3. Four chained 16x16x4 WMMAs give K=16; D rows 0..7 = subbands
// of 16 blocks (set0), rows 8..15 = subbands of the next 16 blocks (set1).

typedef float v2f __attribute__((ext_vector_type(2)));
typedef float v8f __attribute__((ext_vector_type(8)));

#define HAAR_C 0.35355339059327373f   // (1/sqrt(2))^3

__device__ __forceinline__ float haarF(int s, int k) {
    return (__builtin_popcount(s & k) & 1) ? -HAAR_C : HAAR_C;
}

__global__ __launch_bounds__(256)
void dwt3d_haar_wmma(const float* __restrict__ x, float* __restrict__ out) {
    // Fixed problem dims from the reference.
    constexpr int C  = 16;
    constexpr int D  = 64, H  = 128, W  = 128;
    constexpr int Dh = 32, Hh = 64,  Wh = 64;
    constexpr int SUBBAND_STRIDE = 2 * C * Dh * Hh * Wh;  // 4,194,304 floats
    constexpr int IN_PLANE = H * W;                        // depth stride

    const int tid  = threadIdx.x;
    const int lane = tid & 31;
    const int wid  = blockIdx.x * 8 + (tid >> 5);          // 0 .. 65535

    // wid -> (b, c, d, h): one wave per output row (Wh=64 outputs per subband)
    const int h  = wid & 63;
    const int d  = (wid >> 6) & 31;
    const int cc = (wid >> 11) & 15;
    const int bb = wid >> 15;

    // ---------------- A operand (constant per lane) ----------------
    // 16x4 f32 A layout: VGPR0 = K=(lane<16?0:2), VGPR1 = K+1; M = lane&15.
    const int Mlo = lane & 15;
    const int K0  = (lane >> 4) << 1;          // 0 or 2
    const bool top = (Mlo < 8);                // rows 0..7 = set0 filter, 8..15 = set1
    const int  s0  = top ? Mlo : (Mlo - 8);
    const float f0 = haarF(s0, K0);
    const float f1 = haarF(s0, K0 + 1);
    const float f4 = haarF(s0, K0 + 4);
    const float f5 = haarF(s0, K0 + 5);
    // Block-diagonal slices: WMMA0/1 act on rows 0..7 (k=0..3 / 4..7),
    // WMMA2/3 on rows 8..15 (k=0..3 / 4..7 of set1 blocks).
    const v2f a0 = { top ? f0 : 0.0f, top ? f1 : 0.0f };
    const v2f a1 = { top ? f4 : 0.0f, top ? f5 : 0.0f };
    const v2f a2 = { top ? 0.0f : f0, top ? 0.0f : f1 };
    const v2f a3 = { top ? 0.0f : f4, top ? 0.0f : f5 };

    // ---------------- B addressing ----------------
    // B 4x16 layout: VGPR0 = K=(lane<16?0:2), VGPR1 = K+1; N = lane&15.
    // k = 4p + 2q + r  =>  lanes 0..15 carry row 2h (q=0), lanes 16..31 row 2h+1.
    const int n      = lane & 15;
    const int rowsel = lane >> 4;
    const float* base = x + (((bb * C + cc) * D + 2 * d) * H + (2 * h + rowsel)) * W;

    const int outRowBase = (((bb * C + cc) * Dh + d) * Hh + h) * Wh;

    #pragma unroll
    for (int half = 0; half < 2; ++half) {
        const int w0 = half * 32;            // output-w base for this group of 32 blocks
        const int wx = 2 * w0 + 2 * n;       // input-w for set0, lane n

        // Single-touch data: non-temporal loads (each is 2x contiguous 128B runs).
        const v2f b0 = __builtin_nontemporal_load((const v2f*)(base + wx));                  // depth 2d,   set0
        const v2f b1 = __builtin_nontemporal_load((const v2f*)(base + IN_PLANE + wx));       // depth 2d+1, set0
        const v2f b2 = __builtin_nontemporal_load((const v2f*)(base + wx + 32));             // depth 2d,   set1
        const v2f b3 = __builtin_nontemporal_load((const v2f*)(base + IN_PLANE + wx + 32));  // depth 2d+1, set1

        v8f acc = {};
        acc = __builtin_amdgcn_wmma_f32_16x16x4_f32(false, a0, false, b0, (short)0, acc, false, false);
        acc = __builtin_amdgcn_wmma_f32_16x16x4_f32(false, a1, false, b1, (short)0, acc, false, false);
        acc = __builtin_amdgcn_wmma_f32_16x16x4_f32(false, a2, false, b2, (short)0, acc, false, false);
        acc = __builtin_amdgcn_wmma_f32_16x16x4_f32(false, a3, false, b3, (short)0, acc, false, false);

        // D VGPR v = subband v, contiguous w0..w0+31 across the wave (lanes 16..31
        // carry M=v+8 = subband v of set1 at w0+16+(lane-16)  =>  w = w0 + lane).
        const int o = outRowBase + w0 + lane;
        #pragma unroll
        for (int s = 0; s < 8; ++s) {
            __builtin_nontemporal_store(acc[s], out + s * SUBBAND_STRIDE + o);
        }
    }
}

extern "C" void kernel_launch(void* const* d_in, const int* in_sizes, int n_in,
                              void* d_out, int out_size, void* d_ws, size_t ws_size,
                              hipStream_t stream) {
    (void)in_sizes; (void)n_in; (void)out_size; (void)d_ws; (void)ws_size;
    const float* x = (const float*)d_in[0];
    float* out = (float*)d_out;
    // 65536 waves total: B*C*Dh*Hh = 2*16*32*64 rows, 8 waves (256 threads) per block.
    dim3 grid(8192), block(256);
    hipLaunchKernelGGL(dwt3d_haar_wmma, grid, block, 0, stream, x, out);
}